// DMax_34187939676516
// MI455X (gfx1250) — compile-verified
//
#include <hip/hip_runtime.h>

// Segmented ragged max: out[seg][f] = max over rows [start_seg, end_seg - 2) of in[row][f]
// in:   [TOTAL, 64] float32, sizes:[NSEG] int32 (each >= 3), out: [NSEG, 64] float32
//
// Memory-bound: 513MB / 23.3TB/s ~ 22us floor. Max-reduction => no WMMA; hot path is
// direct NT global_load_b128 streaming (LDS staging would add a round-trip with no reuse).
// One 256-thread block (8 wave32) per segment, 8-deep load pipeline, LDS tree reduction,
// final result shipped LDS->global with global_store_async_from_lds_b128 (ASYNCcnt path).

typedef float v4f __attribute__((ext_vector_type(4)));
typedef int   v4i __attribute__((ext_vector_type(4)));

#define NFEAT 64

// ---------------------------------------------------------------------------
// Kernel 1: inclusive prefix sum of sizes -> ends[], single block, n <= 4096
// ---------------------------------------------------------------------------
__global__ __launch_bounds__(1024)
void DMax_scan_sizes(const int* __restrict__ sizes, int* __restrict__ ends, int n) {
    __shared__ int ssum[1024];
    const int t    = threadIdx.x;
    const int base = t << 2;

    v4i s = (v4i){0, 0, 0, 0};
    if (base + 3 < n) {
        s = *(const v4i*)(sizes + base);          // common path: single b128 load
    } else {
        if (base + 0 < n) s.x = sizes[base + 0];
        if (base + 1 < n) s.y = sizes[base + 1];
        if (base + 2 < n) s.z = sizes[base + 2];
    }

    const int e0 = s.x;
    const int e1 = e0 + s.y;
    const int e2 = e1 + s.z;
    const int e3 = e2 + s.w;

    ssum[t] = e3;
    __syncthreads();

    // Hillis-Steele inclusive scan over 1024 per-thread totals
    for (int off = 1; off < 1024; off <<= 1) {
        int v   = ssum[t];
        int add = (t >= off) ? ssum[t - off] : 0;
        __syncthreads();
        ssum[t] = v + add;
        __syncthreads();
    }

    const int prev = (t > 0) ? ssum[t - 1] : 0;
    if (base + 3 < n) {
        v4i e = (v4i){prev + e0, prev + e1, prev + e2, prev + e3};
        *(v4i*)(ends + base) = e;
    } else {
        if (base + 0 < n) ends[base + 0] = prev + e0;
        if (base + 1 < n) ends[base + 1] = prev + e1;
        if (base + 2 < n) ends[base + 2] = prev + e2;
    }
}

// ---------------------------------------------------------------------------
// Kernel 2: one block per segment. 16 float4-lanes x 16 row-lanes = 256 thr.
// ---------------------------------------------------------------------------
__global__ __launch_bounds__(256)
void DMax_seg_max(const float* __restrict__ in,
                  const int*   __restrict__ sizes,
                  const int*   __restrict__ ends,
                  float*       __restrict__ out) {
    const int seg       = blockIdx.x;
    const int end       = ends[seg];
    const int start     = end - sizes[seg];
    const int lastValid = end - 2;              // rows [start, end-2) are reduced
    const int tid       = threadIdx.x;
    const int f4        = (tid & 15) << 2;      // float4 column within the 64-wide row
    const int rl        = tid >> 4;             // row lane 0..15

    const float NEG_INF = -__builtin_inff();
    v4f acc = (v4f){NEG_INF, NEG_INF, NEG_INF, NEG_INF};

    const float* colBase = in + (size_t)f4;

    // Main loop: 8 independent NT b128 loads in flight per thread (4KB/wave),
    // then the (cheap) max chain. Wave lanes cover 512B contiguous per load.
    int row = start + rl;
    for (; row + 16 * 7 < lastValid; row += 16 * 8) {
        v4f v[8];
        #pragma unroll
        for (int u = 0; u < 8; ++u) {
            v[u] = __builtin_nontemporal_load(
                (const v4f*)(colBase + (size_t)(row + 16 * u) * NFEAT));
        }
        #pragma unroll
        for (int u = 0; u < 8; ++u) {
            acc.x = fmaxf(acc.x, v[u].x);
            acc.y = fmaxf(acc.y, v[u].y);
            acc.z = fmaxf(acc.z, v[u].z);
            acc.w = fmaxf(acc.w, v[u].w);
        }
    }
    for (; row < lastValid; row += 16) {
        v4f v = __builtin_nontemporal_load((const v4f*)(colBase + (size_t)row * NFEAT));
        acc.x = fmaxf(acc.x, v.x);
        acc.y = fmaxf(acc.y, v.y);
        acc.z = fmaxf(acc.z, v.z);
        acc.w = fmaxf(acc.w, v.w);
    }

    // LDS tree reduction across the 16 row-lanes (same low-4 bits == same f4).
    __shared__ v4f red[256];
    red[tid] = acc;
    __syncthreads();

    #pragma unroll
    for (int off = 128; off >= 16; off >>= 1) {
        if (tid < off) {
            v4f a = red[tid];
            v4f b = red[tid + off];
            a.x = fmaxf(a.x, b.x);
            a.y = fmaxf(a.y, b.y);
            a.z = fmaxf(a.z, b.z);
            a.w = fmaxf(a.w, b.w);
            red[tid] = a;
        }
        __syncthreads();
    }

    // red[0..15] now hold the segment max (f4 = tid*4). The result already
    // lives in LDS, so ship it LDS->global directly with the CDNA5 async path
    // (global_store_async_from_lds_b128, tracked by ASYNCcnt) instead of a
    // ds_load -> VGPR -> global_store round trip.
    if (tid < 16) {
        float* dst = out + (size_t)seg * NFEAT + (tid << 2);
#if defined(__gfx1250__)
        typedef __attribute__((address_space(3))) v4f* lds_v4f_p;
        lds_v4f_p lsrc = (lds_v4f_p)&red[tid];
        asm volatile(
            "s_wait_dscnt 0x0\n\t"
            "global_store_async_from_lds_b128 %0, %1, off\n\t"
            "s_wait_asynccnt 0x0"
            :
            : "v"(dst), "v"(lsrc)
            : "memory");
#else
        *(v4f*)dst = red[tid];
#endif
    }
}

// ---------------------------------------------------------------------------
extern "C" void kernel_launch(void* const* d_in, const int* in_sizes, int n_in,
                              void* d_out, int out_size, void* d_ws, size_t ws_size,
                              hipStream_t stream) {
    const float* input = (const float*)d_in[0];   // [TOTAL, 64] f32
    const int*   sizes = (const int*)d_in[1];     // [NSEG] i32
    const int    n_seg = in_sizes[1];             // 4096

    int* ends = (int*)d_ws;                       // n_seg ints of scratch

    DMax_scan_sizes<<<1, 1024, 0, stream>>>(sizes, ends, n_seg);
    DMax_seg_max<<<n_seg, 256, 0, stream>>>(input, sizes, ends, (float*)d_out);
}